// DPStack_1305670058331
// MI455X (gfx1250) — compile-verified
//
#include <hip/hip_runtime.h>
#include <math.h>

typedef float v2f __attribute__((ext_vector_type(2)));
typedef float v8f __attribute__((ext_vector_type(8)));

#define S64 64
#define H512 512
#define V32K 32000
#define NPAIR 2016          // 64*63/2
#define NROWS 2017          // pairs + init row
#define NROWS_PAD 2032      // multiple of 16
#define SENT_LEN 63
#define LDS_STRIDE 516      // 512 + 4 pad -> conflict-free strided reads
#define NCHUNK 4            // vocab split: 4 * 500 tiles of 16 = 32000
#define TILES_PER_CHUNK 500
#define GROUPS_PER_CHUNK 125  // groups of 4 n-tiles

// ---------- helpers ----------
__device__ __forceinline__ void pair_ij(int p, int& i, int& j) {
    int ii = 0, off = 0;
    while (off + (S64 - 1 - ii) <= p) { off += S64 - 1 - ii; ++ii; }
    i = ii;
    j = ii + 1 + (p - off);
}

__device__ __forceinline__ float logsigf(float x) {
    // log(sigmoid(x)) = min(x,0) - log1p(exp(-|x|))
    return fminf(x, 0.f) - log1pf(expf(-fabsf(x)));
}

// ---------------------------------------------------------------------------
// Kernel A: four GEMMs  out[64,512] = enc[64,512] @ W_half[512,512]
// which: 0->At (Wt1 top) 1->Bt (Wt1 bot) 2->Aw (Ww1 top) 3->Bw (Ww1 bot)
// one wave per 16x16 output tile; f32 WMMA 16x16x4, K=512
// ---------------------------------------------------------------------------
__global__ __launch_bounds__(32) void gemm64_kernel(
    const float* __restrict__ enc, const float* __restrict__ Wt1,
    const float* __restrict__ Ww1, float* __restrict__ At, float* __restrict__ Bt,
    float* __restrict__ Aw, float* __restrict__ Bw)
{
    const int n0 = blockIdx.x * 16;
    const int m0 = blockIdx.y * 16;
    const int which = blockIdx.z;

    const float* W = (which < 2) ? Wt1 : Ww1;
    const float* Wb = W + ((which & 1) ? (H512 * H512) : 0);
    float* out = (which == 0) ? At : (which == 1) ? Bt : (which == 2) ? Aw : Bw;

    const int lane = threadIdx.x;
    const int half = lane >> 4;
    const int lr   = lane & 15;

    v8f acc = {0.f, 0.f, 0.f, 0.f, 0.f, 0.f, 0.f, 0.f};
    for (int k0 = 0; k0 < H512; k0 += 4) {
        const int ka = k0 + 2 * half;
        v2f av;  // A 16x4: lanes0-15 K=k0,k0+1 ; lanes16-31 K=k0+2,k0+3
        const float* ap = enc + (m0 + lr) * H512 + ka;
        av.x = ap[0]; av.y = ap[1];
        v2f bv;  // B 4x16: lanes0-15 K=k0,k0+1 ; lanes16-31 K=k0+2,k0+3
        const float* bp = Wb + ka * H512 + n0 + lr;
        bv.x = bp[0]; bv.y = bp[H512];
        acc = __builtin_amdgcn_wmma_f32_16x16x4_f32(false, av, false, bv,
                                                    (short)0, acc, false, false);
    }
#pragma unroll
    for (int r = 0; r < 8; ++r)
        out[(m0 + r + 8 * half) * H512 + n0 + lr] = acc[r];
}

// ---------------------------------------------------------------------------
// Kernel B: per-pair prep (one wave per row)
//  - re_logit -> LOGRE[i][j], lognre[p]
//  - Hw[p,:] = tanh(Aw[i]+Bw[j]+bw1)   (row 2016 = init row i=j=0; pads zero)
//  - nid[p]
// ---------------------------------------------------------------------------
__global__ __launch_bounds__(256) void pair_prep_kernel(
    const float* __restrict__ At, const float* __restrict__ Bt,
    const float* __restrict__ Aw, const float* __restrict__ Bw,
    const float* __restrict__ bt1, const float* __restrict__ Wt2,
    const float* __restrict__ bt2, const float* __restrict__ bw1,
    const int* __restrict__ word_ids,
    float* __restrict__ LOGRE, float* __restrict__ lognre,
    float* __restrict__ Hw, int* __restrict__ nid)
{
    const int lane = threadIdx.x & 31;
    const int p = blockIdx.x * 8 + (threadIdx.x >> 5);
    if (p >= NROWS_PAD) return;

    if (p >= NROWS) {  // zero padding rows
        for (int t = 0; t < 16; ++t) Hw[p * H512 + t * 32 + lane] = 0.f;
        return;
    }

    int i = 0, j = 0;
    if (p < NPAIR) {
        pair_ij(p, i, j);
        // retention head: dot(tanh(At[i]+Bt[j]+bt1), Wt2) + bt2
        float part = 0.f;
        for (int t = 0; t < 16; ++t) {
            const int c = t * 32 + lane;
            part += tanhf(At[i * H512 + c] + Bt[j * H512 + c] + bt1[c]) * Wt2[c];
        }
        for (int o = 16; o > 0; o >>= 1) part += __shfl_xor(part, o, 32);
        const float x = part + bt2[0];
        if (lane == 0) {
            LOGRE[i * S64 + j] = logsigf(x);
            lognre[p] = logsigf(-x);
            nid[p] = word_ids[min(j + 1, S64 - 1)];
        }
    } else {  // p == 2016: init row uses [enc0, enc0]
        if (lane == 0) nid[p] = word_ids[1];
    }

    const int ii = (p == NPAIR) ? 0 : i;
    const int jj = (p == NPAIR) ? 0 : j;
    for (int t = 0; t < 16; ++t) {
        const int c = t * 32 + lane;
        Hw[p * H512 + c] = tanhf(Aw[ii * H512 + c] + Bw[jj * H512 + c] + bw1[c]);
    }
}

// ---------------------------------------------------------------------------
// Kernel C: logits = Hw[16 rows] @ Ww2 + bw2 for one vocab chunk (8000 cols),
// fused streaming logsumexp (partial per chunk) + gather of logit at nid[row].
// grid = (127 row tiles, 4 vocab chunks); 8 waves stripe 125 groups of 4
// n-tiles; one shared A-fragment feeds 4 WMMAs (4 independent acc chains).
// ---------------------------------------------------------------------------
__global__ __launch_bounds__(256) void logits_lse_kernel(
    const float* __restrict__ Hw, const float* __restrict__ Ww2,
    const float* __restrict__ bw2, const int* __restrict__ nid,
    float* __restrict__ pm, float* __restrict__ ps,
    float* __restrict__ gathered)
{
    __shared__ float sH[16 * LDS_STRIDE];
    __shared__ float red_m[8][16];
    __shared__ float red_s[8][16];
    __shared__ int snid[16];

    const int rt = blockIdx.x;     // 0..126
    const int chunk = blockIdx.y;  // 0..3
    for (int e = threadIdx.x; e < 16 * H512; e += 256) {
        const int r = e >> 9, c = e & 511;
        sH[r * LDS_STRIDE + c] = Hw[(rt * 16 + r) * H512 + c];
    }
    if (threadIdx.x < 16) {
        const int row = rt * 16 + threadIdx.x;
        snid[threadIdx.x] = (row < NROWS) ? nid[row] : -1;
    }
    __syncthreads();

    const int lane = threadIdx.x & 31;
    const int wid  = threadIdx.x >> 5;
    const int half = lane >> 4;
    const int lr   = lane & 15;

    float m8[8], s8[8];
#pragma unroll
    for (int r = 0; r < 8; ++r) { m8[r] = -INFINITY; s8[r] = 0.f; }

    for (int g = wid; g < GROUPS_PER_CHUNK; g += 8) {
        const int c0 = chunk * (TILES_PER_CHUNK * 16) + g * 64 + lr;  // lane col
        v8f a0 = {0.f, 0.f, 0.f, 0.f, 0.f, 0.f, 0.f, 0.f};
        v8f a1 = a0, a2 = a0, a3 = a0;
        for (int k0 = 0; k0 < H512; k0 += 4) {
            const int ka = k0 + 2 * half;
            const v2f av = *(const v2f*)&sH[lr * LDS_STRIDE + ka];
            const float* bp = Ww2 + (size_t)ka * V32K + c0;
            v2f b0, b1, b2, b3;
            b0.x = bp[0];  b0.y = bp[V32K];
            b1.x = bp[16]; b1.y = bp[V32K + 16];
            b2.x = bp[32]; b2.y = bp[V32K + 32];
            b3.x = bp[48]; b3.y = bp[V32K + 48];
            a0 = __builtin_amdgcn_wmma_f32_16x16x4_f32(false, av, false, b0,
                                                       (short)0, a0, false, false);
            a1 = __builtin_amdgcn_wmma_f32_16x16x4_f32(false, av, false, b1,
                                                       (short)0, a1, false, false);
            a2 = __builtin_amdgcn_wmma_f32_16x16x4_f32(false, av, false, b2,
                                                       (short)0, a2, false, false);
            a3 = __builtin_amdgcn_wmma_f32_16x16x4_f32(false, av, false, b3,
                                                       (short)0, a3, false, false);
        }
#pragma unroll
        for (int sub = 0; sub < 4; ++sub) {
            const v8f acc = (sub == 0) ? a0 : (sub == 1) ? a1 : (sub == 2) ? a2 : a3;
            const int col = c0 + sub * 16;
            const float bias = bw2[col];
#pragma unroll
            for (int r = 0; r < 8; ++r) {
                const float x = acc[r] + bias;      // logit(row = r+8*half, col)
                const int rl = r + 8 * half;
                if (col == snid[rl]) gathered[rt * 16 + rl] = x;  // unique writer
                const float nm = fmaxf(m8[r], x);   // streaming logsumexp
                s8[r] = s8[r] * expf(m8[r] - nm) + expf(x - nm);
                m8[r] = nm;
            }
        }
    }

    // reduce across the 16 lanes of each half (xor of bits 0..3 stays in half)
#pragma unroll
    for (int r = 0; r < 8; ++r) {
        float m = m8[r], s = s8[r];
        for (int o = 1; o < 16; o <<= 1) {
            const float om = __shfl_xor(m, o, 32);
            const float os = __shfl_xor(s, o, 32);
            const float nm = fmaxf(m, om);
            s = s * expf(m - nm) + os * expf(om - nm);
            m = nm;
        }
        if (lr == 0) { red_m[wid][r + 8 * half] = m; red_s[wid][r + 8 * half] = s; }
    }
    __syncthreads();

    if (threadIdx.x < 16) {
        float m = -INFINITY, s = 0.f;
        for (int w = 0; w < 8; ++w) {
            const float om = red_m[w][threadIdx.x], os = red_s[w][threadIdx.x];
            const float nm = fmaxf(m, om);
            s = s * expf(m - nm) + os * expf(om - nm);
            m = nm;
        }
        const int row = rt * 16 + threadIdx.x;
        if (row < NROWS) {
            pm[chunk * NROWS_PAD + row] = m;
            ps[chunk * NROWS_PAD + row] = s;
        }
    }
}

// ---------------------------------------------------------------------------
// Kernel C2: combine the 4 per-chunk partial (max,sum) into lse[row].
// ---------------------------------------------------------------------------
__global__ __launch_bounds__(256) void lse_combine_kernel(
    const float* __restrict__ pm, const float* __restrict__ ps,
    float* __restrict__ lse)
{
    const int row = blockIdx.x * 256 + threadIdx.x;
    if (row >= NROWS) return;
    float m = -INFINITY, s = 0.f;
#pragma unroll
    for (int c = 0; c < NCHUNK; ++c) {
        const float om = pm[c * NROWS_PAD + row], os = ps[c * NROWS_PAD + row];
        const float nm = fmaxf(m, om);
        s = s * expf(m - nm) + os * expf(om - nm);
        m = nm;
    }
    lse[row] = m + logf(s);
}

// ---------------------------------------------------------------------------
// Kernel D: inside-algorithm DP, single 1024-thread block, gap-serial.
// ---------------------------------------------------------------------------
__global__ __launch_bounds__(1024) void dp_kernel(
    const float* __restrict__ LOGRE, const float* __restrict__ lognre,
    const float* __restrict__ gathered, const float* __restrict__ lse,
    float* __restrict__ table, float* __restrict__ out)
{
    const int tid = threadIdx.x;
    for (int e = tid; e < S64 * S64 * S64; e += 1024) table[e] = -INFINITY;
    __syncthreads();

    // base cases
    for (int p = tid; p < NROWS; p += 1024) {
        if (p == NPAIR) {
            table[(0 * S64 + 0) * S64 + 1] = gathered[NPAIR] - lse[NPAIR];
        } else {
            int i, j; pair_ij(p, i, j);
            if (j <= SENT_LEN - 1) {
                const float wt = gathered[p] - lse[p];
                table[(i * S64 + j) * S64 + (j + 1)] = lognre[p] + wt;
            }
        }
    }
    __syncthreads();

    for (int gap = 2; gap <= SENT_LEN; ++gap) {
        const int imax = SENT_LEN - gap;            // i in [0, imax]
        const int total = (imax >= 1) ? 1 + imax * (imax + 1) / 2 : 1;
        for (int cell = tid; cell < total; cell += 1024) {
            int i, l;
            if (cell == 0) { i = 0; l = 0; }
            else {
                const int c = cell - 1;             // rows i>=1, li = i cells
                i = 1;
                while (i * (i + 1) / 2 <= c) ++i;
                l = c - (i - 1) * i / 2;
            }
            const int j = i + gap;
            const float* trow = table + (l * S64 + i) * S64;
            float m = -INFINITY;
            for (int k = i + 1; k < j; ++k) {
                const float v = trow[k] + table[(i * S64 + k) * S64 + j] +
                                LOGRE[k * S64 + j];
                m = fmaxf(m, v);
            }
            float res = -INFINITY;
            if (m > -INFINITY) {
                float s = 0.f;
                for (int k = i + 1; k < j; ++k) {
                    const float v = trow[k] + table[(i * S64 + k) * S64 + j] +
                                    LOGRE[k * S64 + j];
                    s += expf(v - m);
                }
                res = m + logf(s);
            }
            table[(l * S64 + i) * S64 + j] = res;
        }
        __syncthreads();
    }
    if (tid == 0) out[0] = table[(0 * S64 + 0) * S64 + SENT_LEN];
}

// ---------------------------------------------------------------------------
extern "C" void kernel_launch(void* const* d_in, const int* in_sizes, int n_in,
                              void* d_out, int out_size, void* d_ws, size_t ws_size,
                              hipStream_t stream) {
    const float* enc  = (const float*)d_in[0];   // [64,512]
    const int*   wid  = (const int*)  d_in[1];   // [64]
    const float* Wt1  = (const float*)d_in[2];   // [1024,512]
    const float* bt1  = (const float*)d_in[3];   // [512]
    const float* Wt2  = (const float*)d_in[4];   // [512,1]
    const float* bt2  = (const float*)d_in[5];   // [1]
    const float* Ww1  = (const float*)d_in[6];   // [1024,512]
    const float* bw1  = (const float*)d_in[7];   // [512]
    const float* Ww2  = (const float*)d_in[8];   // [512,32000]
    const float* bw2  = (const float*)d_in[9];   // [32000]
    float* out = (float*)d_out;

    // workspace layout (floats)
    float* ws = (float*)d_ws;
    float* At      = ws;                       // 32768
    float* Bt      = At + 64 * 512;
    float* Aw      = Bt + 64 * 512;
    float* Bw      = Aw + 64 * 512;
    float* Hw      = Bw + 64 * 512;            // 2032*512
    float* LOGRE   = Hw + NROWS_PAD * 512;     // 4096
    float* lognre  = LOGRE + 64 * 64;          // 2048 (padded)
    float* lse     = lognre + 2048;            // 2048
    float* gathered= lse + 2048;               // 2048
    float* pm      = gathered + 2048;          // 4*2032 = 8128
    float* ps      = pm + NCHUNK * NROWS_PAD;  // 8128
    int*   nid     = (int*)(ps + NCHUNK * NROWS_PAD);  // 2048
    float* table   = (float*)(nid + 2048);     // 64^3

    gemm64_kernel<<<dim3(32, 4, 4), 32, 0, stream>>>(enc, Wt1, Ww1, At, Bt, Aw, Bw);
    pair_prep_kernel<<<NROWS_PAD / 8, 256, 0, stream>>>(
        At, Bt, Aw, Bw, bt1, Wt2, bt2, bw1, wid, LOGRE, lognre, Hw, nid);
    logits_lse_kernel<<<dim3(NROWS_PAD / 16, NCHUNK), 256, 0, stream>>>(
        Hw, Ww2, bw2, nid, pm, ps, gathered);
    lse_combine_kernel<<<(NROWS + 255) / 256, 256, 0, stream>>>(pm, ps, lse);
    dp_kernel<<<1, 1024, 0, stream>>>(LOGRE, lognre, gathered, lse, table, out);
}